// GateGATLayer_45887430591131
// MI455X (gfx1250) — compile-verified
//
#include <hip/hip_runtime.h>

typedef float v2f __attribute__((ext_vector_type(2)));
typedef float v8f __attribute__((ext_vector_type(8)));

#define IN_DIM   128
#define OUT_DIM  64
#define NEG_SLOPE 0.01f
#define PAIR_STRIDE 80   // v2f per K-pair row; 160 floats -> rows r, r+1 hit disjoint 32-bank halves

// ---------------------------------------------------------------------------
// z = h @ fc_w^T :  (N x 128) * (64 x 128)^T -> (N x 64), f32 WMMA 16x16x4
// One wave32 computes a 16(M) x 64(N) tile with 4 accumulators.
// fc_w is staged in LDS as interleaved K-pairs so each B fragment is a single
// aligned ds_load_b64 (no VGPR repacking, conflict-free banking).
// ---------------------------------------------------------------------------
__global__ __launch_bounds__(256)
void gemm_z_kernel(const float* __restrict__ h, const float* __restrict__ w,
                   float* __restrict__ z, int N)
{
    // lwP[r * 80 + o] = { fc_w[o][2r], fc_w[o][2r+1] },  r = 0..63, o = 0..63
    __shared__ v2f lwP[(IN_DIM / 2) * PAIR_STRIDE];      // 40 KB
    {
        float* lw = (float*)lwP;
        for (int i = threadIdx.x; i < OUT_DIM * IN_DIM; i += blockDim.x) {
            int o = i >> 7;          // row of fc_w  (0..63)
            int k = i & 127;         // col of fc_w  (0..127)
            lw[((k >> 1) * PAIR_STRIDE + o) * 2 + (k & 1)] = w[i];
        }
    }
    __syncthreads();

    const int wave = threadIdx.x >> 5;
    const int lane = threadIdx.x & 31;
    const long tile = (long)blockIdx.x * 8 + wave;
    const long rowBase = tile * 16;
    if (rowBase >= N) return;                    // wave-uniform: EXEC stays all-ones

    const int m     = lane & 15;
    const int khalf = (lane >> 4) << 1;          // 0 for lanes 0-15, 2 for lanes 16-31
    long arow = rowBase + m;
    if (arow >= N) arow = N - 1;                 // clamp instead of predicating (EXEC all-ones)
    const float* __restrict__ hrow = h + arow * IN_DIM;

    v8f acc0 = {}, acc1 = {}, acc2 = {}, acc3 = {};
    const int o0 = lane & 15;

    #pragma unroll
    for (int k = 0; k < IN_DIM; k += 4) {
        // A fragment (16x4 f32): lane<16 -> K=k,k+1 ; lane>=16 -> K=k+2,k+3
        v2f a = *(const v2f*)(hrow + k + khalf);

        // B fragments (4x16 f32): lane holds (B[kb][o], B[kb+1][o]), kb = k+khalf
        const v2f* bp = &lwP[((k + khalf) >> 1) * PAIR_STRIDE + o0];
        v2f b0 = bp[ 0];
        v2f b1 = bp[16];
        v2f b2 = bp[32];
        v2f b3 = bp[48];

        acc0 = __builtin_amdgcn_wmma_f32_16x16x4_f32(false, a, false, b0, (short)0, acc0, false, false);
        acc1 = __builtin_amdgcn_wmma_f32_16x16x4_f32(false, a, false, b1, (short)0, acc1, false, false);
        acc2 = __builtin_amdgcn_wmma_f32_16x16x4_f32(false, a, false, b2, (short)0, acc2, false, false);
        acc3 = __builtin_amdgcn_wmma_f32_16x16x4_f32(false, a, false, b3, (short)0, acc3, false, false);
    }

    // D layout: VGPR r -> lanes 0-15: M=r, lanes 16-31: M=r+8; N = lane&15 within tile
    const int col  = lane & 15;
    const int rofs = (lane >> 4) << 3;           // 0 or 8
    #pragma unroll
    for (int r = 0; r < 8; ++r) {
        long zr = rowBase + r + rofs;
        if (zr < N) {
            float* zp = z + zr * OUT_DIM + col;
            zp[ 0] = acc0[r];
            zp[16] = acc1[r];
            zp[32] = acc2[r];
            zp[48] = acc3[r];
        }
    }
}

// ---------------------------------------------------------------------------
// Per-edge score: s = leakyrelu(z[src].a_src + z[dst].a_dst) * gate
// One wave32 per edge: coalesced 256B row gathers, shfl reduction.
// Segment-max via monotonic-uint atomicMax (identity key == 0).
// ---------------------------------------------------------------------------
__global__ __launch_bounds__(256)
void edge_score_kernel(const float* __restrict__ z, const float* __restrict__ gate,
                       const float* __restrict__ attn_w,
                       const int* __restrict__ src, const int* __restrict__ dst,
                       float* __restrict__ s_out, unsigned int* __restrict__ mkey, int E)
{
    const int lane = threadIdx.x & 31;
    const long e = ((long)blockIdx.x * blockDim.x + threadIdx.x) >> 5;
    if (e >= E) return;
    const int si = src[e];
    const int di = dst[e];

    v2f vs = *(const v2f*)(z + (long)si * OUT_DIM + lane * 2);
    v2f vd = *(const v2f*)(z + (long)di * OUT_DIM + lane * 2);
    v2f as = *(const v2f*)(attn_w + lane * 2);
    v2f ad = *(const v2f*)(attn_w + OUT_DIM + lane * 2);

    float p = vs.x * as.x + vs.y * as.y + vd.x * ad.x + vd.y * ad.y;
    #pragma unroll
    for (int off = 16; off > 0; off >>= 1)
        p += __shfl_xor(p, off, 32);

    if (lane == 0) {
        float ev = (p >= 0.f) ? p : NEG_SLOPE * p;
        float s  = ev * gate[e];
        s_out[e] = s;
        unsigned u   = __float_as_uint(s);
        unsigned key = (u & 0x80000000u) ? ~u : (u | 0x80000000u);  // order-preserving
        atomicMax(&mkey[di], key);
    }
}

// ---------------------------------------------------------------------------
// Scatter: ex = exp(s - m[dst]); denom[dst]+=ex; num[dst] += ex * z[src]
// One wave32 per edge; num accumulates directly in d_out (L2-resident atomics).
// ---------------------------------------------------------------------------
__global__ __launch_bounds__(256)
void edge_scatter_kernel(const float* __restrict__ z,
                         const int* __restrict__ src, const int* __restrict__ dst,
                         const float* __restrict__ s_in, const unsigned int* __restrict__ mkey,
                         float* __restrict__ denom, float* __restrict__ num, int E)
{
    const int lane = threadIdx.x & 31;
    const long e = ((long)blockIdx.x * blockDim.x + threadIdx.x) >> 5;
    if (e >= E) return;
    const int si = src[e];
    const int di = dst[e];

    const unsigned key = mkey[di];
    const unsigned u   = (key & 0x80000000u) ? (key & 0x7fffffffu) : ~key;  // decode
    const float    m   = __uint_as_float(u);
    const float    ex  = __expf(s_in[e] - m);

    if (lane == 0) atomicAdd(&denom[di], ex);

    v2f vz = *(const v2f*)(z + (long)si * OUT_DIM + lane * 2);
    float* np = num + (long)di * OUT_DIM + lane * 2;
    atomicAdd(np + 0, ex * vz.x);
    atomicAdd(np + 1, ex * vz.y);
}

// ---------------------------------------------------------------------------
// h_out = num / where(denom > 0, denom, 1)   (in place on d_out)
// ---------------------------------------------------------------------------
__global__ __launch_bounds__(256)
void finalize_kernel(float* __restrict__ out, const float* __restrict__ denom, long total)
{
    long i = (long)blockIdx.x * blockDim.x + threadIdx.x;
    if (i >= total) return;
    float d = denom[i >> 6];
    out[i] = out[i] / ((d > 0.f) ? d : 1.f);
}

// ---------------------------------------------------------------------------
extern "C" void kernel_launch(void* const* d_in, const int* in_sizes, int n_in,
                              void* d_out, int out_size, void* d_ws, size_t ws_size,
                              hipStream_t stream)
{
    const float* h      = (const float*)d_in[0];
    const float* gate   = (const float*)d_in[1];
    const float* fc_w   = (const float*)d_in[2];
    const float* attn_w = (const float*)d_in[3];
    const int*   src    = (const int*)d_in[4];
    const int*   dst    = (const int*)d_in[5];

    const int N = in_sizes[0] / IN_DIM;
    const int E = in_sizes[4];
    float* out = (float*)d_out;

    // Workspace carve-out (256B aligned): z, s, mkey, denom
    char* ws = (char*)d_ws;
    size_t off = 0;
    float* z = (float*)(ws + off);
    off = (off + (size_t)N * OUT_DIM * sizeof(float) + 255) & ~(size_t)255;
    float* s = (float*)(ws + off);
    off = (off + (size_t)E * sizeof(float) + 255) & ~(size_t)255;
    unsigned int* mkey = (unsigned int*)(ws + off);
    off = (off + (size_t)N * sizeof(unsigned int) + 255) & ~(size_t)255;
    float* denom = (float*)(ws + off);

    // Re-initialize accumulators every call (d_out is poisoned; determinism required)
    hipMemsetAsync(out,   0, (size_t)N * OUT_DIM * sizeof(float), stream);
    hipMemsetAsync(mkey,  0, (size_t)N * sizeof(unsigned int), stream);
    hipMemsetAsync(denom, 0, (size_t)N * sizeof(float), stream);

    // 1) GEMM: z = h @ fc_w^T
    const int tiles   = (N + 15) / 16;
    const int gblocks = (tiles + 7) / 8;
    gemm_z_kernel<<<gblocks, 256, 0, stream>>>(h, fc_w, z, N);

    // 2) + 3) edge phases: one wave32 per edge
    const long ethreads = (long)E * 32;
    const int  eblocks  = (int)((ethreads + 255) / 256);
    edge_score_kernel<<<eblocks, 256, 0, stream>>>(z, gate, attn_w, src, dst, s, mkey, E);
    edge_scatter_kernel<<<eblocks, 256, 0, stream>>>(z, src, dst, s, mkey, denom, out, E);

    // 4) finalize
    const long total = (long)N * OUT_DIM;
    finalize_kernel<<<(int)((total + 255) / 256), 256, 0, stream>>>(out, denom, total);
}